// SpikingNeuralNetwork_50964081935166
// MI455X (gfx1250) — compile-verified
//
#include <hip/hip_runtime.h>

typedef float v2f __attribute__((ext_vector_type(2)));
typedef float v8f __attribute__((ext_vector_type(8)));

#define B_DIM   8192
#define IN_DIM  4096
#define N_DIM   256
#define OUT_DIM 512
#define P_SPLIT 16

// ---------------------------------------------------------------------------
// Kernel 1: column-sum of x [8192 x 4096] via V_WMMA_F32_16X16X4_F32.
// D = ones(16x4) * B(4x16) + C  => every row of D = colsum of the 4x16 chunk.
// Each wave owns a 16-column stripe and 512 rows; writes one partial vector.
// ---------------------------------------------------------------------------
__global__ void __launch_bounds__(256) k_colsum_wmma(const float* __restrict__ x,
                                                     float* __restrict__ partial) {
    const int cb   = blockIdx.x & 31;    // 32 column blocks of 128 cols
    const int p    = blockIdx.x >> 5;    // 16 row splits
    const int lane = threadIdx.x & 31;
    const int wid  = threadIdx.x >> 5;   // 8 waves per block
    const int col  = cb * 128 + wid * 16 + (lane & 15);
    const int rhalf = (lane >> 4) & 1;   // lanes 16..31 take K=2,3 rows
    const int r0 = p * (B_DIM / P_SPLIT);
    const int r1 = r0 + (B_DIM / P_SPLIT);

    v2f a; a.x = 1.0f; a.y = 1.0f;       // all-ones A matrix (16x4)
    v8f acc = {};
    const float* colp = x + (size_t)col;

    for (int r = r0; r < r1; r += 4) {
        const int row = r + rhalf * 2;
        v2f b;
        b.x = colp[(size_t)row * IN_DIM];
        b.y = colp[(size_t)(row + 1) * IN_DIM];
        __builtin_prefetch(&colp[(size_t)(row + 8) * IN_DIM], 0, 1); // global_prefetch_b8
        acc = __builtin_amdgcn_wmma_f32_16x16x4_f32(false, a, false, b,
                                                    (short)0, acc, false, false);
    }
    // D row M=0 lives in acc[0], lanes 0..15, N = lane
    if (lane < 16) partial[p * IN_DIM + col] = acc[0];
}

// ---------------------------------------------------------------------------
// Kernel 2: transpose hidden weights so spike-column updates read contiguous.
// ---------------------------------------------------------------------------
__global__ void k_transpose(const float* __restrict__ W, float* __restrict__ WT) {
    const int idx = blockIdx.x * 256 + threadIdx.x;   // 65536 elements
    const int i = idx >> 8, j = idx & 255;
    WT[j * N_DIM + i] = W[idx];
}

// ---------------------------------------------------------------------------
// Kernel 3: the sequential SNN simulation. One block of 256 threads:
//   all 8 waves: reduce partials -> xmean (LDS), base = xmean @ Win
//   wave 0     : sequential neuron scan (8 neurons/lane in registers)
//   all 8 waves: avg_out = spike_count @ Wout / T
// ---------------------------------------------------------------------------
__global__ void __launch_bounds__(256) k_snn(const float* __restrict__ Win,
                                             const float* __restrict__ W,
                                             const float* __restrict__ Wout,
                                             const int*   __restrict__ tsteps,
                                             const float* __restrict__ partial,
                                             const float* __restrict__ WT,
                                             float* __restrict__ avg) {
    __shared__ float xmean[IN_DIM];
    __shared__ float baseI[N_DIM];
    __shared__ float flag[N_DIM];
    __shared__ float cntf[N_DIM];

    const int t = threadIdx.x;

    // --- reduce row-split partials into the batch-mean of x ---
    for (int c = t; c < IN_DIM; c += 256) {
        float s = 0.f;
        for (int p = 0; p < P_SPLIT; ++p) s += partial[p * IN_DIM + c];
        xmean[c] = s * (1.0f / (float)B_DIM);
    }
    flag[t] = 0.0f;
    __syncthreads();

    // --- base current per neuron: xmean @ Win (coalesced over neurons) ---
    {
        float b = 0.f;
        for (int c = 0; c < IN_DIM; ++c) b += xmean[c] * Win[c * N_DIM + t];
        baseI[t] = b;
    }
    __syncthreads();

    const int T = *tsteps;

    if (t < 32) {                       // wave 0 runs the sequential scan
        const int lane = t;
        float v[8], sp[8], cnt[8], bs[8];
        #pragma unroll
        for (int c = 0; c < 8; ++c) {
            v[c] = 0.f; sp[c] = -1e30f; cnt[c] = 0.f;
            bs[c] = baseI[lane * 8 + c];
        }

        for (int tt = 0; tt < T; ++tt) {
            const float tf = (float)tt;
            float acc[8];
            // suffix term: previous-step flags for j >= i (exact; scan only
            // rewrites flag[j] at j's own turn, after which i>j uses acc updates)
            #pragma unroll
            for (int c = 0; c < 8; ++c) {
                const int i = lane * 8 + c;
                float a = bs[c];
                const float* wr = W + (size_t)i * N_DIM;
                for (int j = i; j < N_DIM; ++j) a += wr[j] * flag[j];
                acc[c] = a;
            }
            // sequential neuron scan, spike broadcast via shfl (wave-uniform)
            for (int jb = 0; jb < 32; ++jb) {
                #pragma unroll
                for (int c = 0; c < 8; ++c) {
                    float spikef = 0.f;
                    if (lane == jb) {
                        const bool  refr  = (tf - sp[c]) < 2.0f;
                        const float vn    = refr ? 0.f : (v[c] + (acc[c] - v[c]) * 0.1f);
                        const bool  spike = (!refr) && (vn >= 1.0f);
                        v[c] = (spike || refr) ? 0.f : vn;
                        if (spike) { sp[c] = tf; cnt[c] += 1.f; spikef = 1.f; }
                        flag[jb * 8 + c] = spikef;
                    }
                    spikef = __shfl(spikef, jb, 32);
                    if (spikef != 0.f) {            // uniform branch: rank-1 update
                        const int j = jb * 8 + c;
                        const float4* cp = (const float4*)(WT + (size_t)j * N_DIM + lane * 8);
                        const float4 w0 = cp[0], w1 = cp[1];
                        const float wv[8] = {w0.x, w0.y, w0.z, w0.w, w1.x, w1.y, w1.z, w1.w};
                        if (lane > jb) {
                            #pragma unroll
                            for (int c2 = 0; c2 < 8; ++c2) acc[c2] += wv[c2];
                        } else if (lane == jb) {
                            #pragma unroll
                            for (int c2 = c + 1; c2 < 8; ++c2) acc[c2] += wv[c2];
                        }
                    }
                }
            }
        }
        #pragma unroll
        for (int c = 0; c < 8; ++c) cntf[lane * 8 + c] = cnt[c];
    }
    __syncthreads();

    // --- avg output: (spike counts @ Wout) / T ---
    const float invT = 1.0f / (float)T;
    for (int o = t; o < OUT_DIM; o += 256) {
        float a = 0.f;
        for (int i = 0; i < N_DIM; ++i) a += cntf[i] * Wout[i * OUT_DIM + o];
        avg[o] = a * invT;
    }
}

// ---------------------------------------------------------------------------
// Kernel 4: broadcast avg[512] to out[8192, 512] with float4 stores.
// ---------------------------------------------------------------------------
__global__ void k_broadcast(const float* __restrict__ avg,
                            float4* __restrict__ out, int n4) {
    const int idx = blockIdx.x * 256 + threadIdx.x;
    if (idx < n4) {
        out[idx] = ((const float4*)avg)[idx & (OUT_DIM / 4 - 1)];
    }
}

extern "C" void kernel_launch(void* const* d_in, const int* in_sizes, int n_in,
                              void* d_out, int out_size, void* d_ws, size_t ws_size,
                              hipStream_t stream) {
    const float* x     = (const float*)d_in[0];
    const float* Win   = (const float*)d_in[1];
    const float* W     = (const float*)d_in[2];
    const float* Wout  = (const float*)d_in[3];
    const int*   tstep = (const int*)d_in[4];

    float* ws      = (float*)d_ws;
    float* partial = ws;                         // 16 * 4096 floats
    float* WT      = ws + P_SPLIT * IN_DIM;      // 65536 floats
    float* avg     = WT + N_DIM * N_DIM;         // 512 floats

    k_colsum_wmma<<<32 * P_SPLIT, 256, 0, stream>>>(x, partial);
    k_transpose<<<N_DIM * N_DIM / 256, 256, 0, stream>>>(W, WT);
    k_snn<<<1, 256, 0, stream>>>(Win, W, Wout, tstep, partial, WT, avg);

    const int n4 = out_size / 4;
    k_broadcast<<<(n4 + 255) / 256, 256, 0, stream>>>(avg, (float4*)d_out, n4);
}